// At_55439437856807
// MI455X (gfx1250) — compile-verified
//
#include <hip/hip_runtime.h>
#include <math.h>
#include <limits.h>
#include <stdint.h>

// Geometry constants (match reference)
#define NB 2
#define NA 360
#define NV 256
#define NU 256
#define NZ 128
#define NY 128
#define NX 128
// dz=dy=dx = 2.0 mm ; dv=du = 2.0 mm ; DSO=1000 ; DSD=1536

#define TILE_X 16
#define TILE_Y 16
#define ZCHUNK 16
#define TILE_CAP 4096   // floats per LDS buffer (16 KB); worst-case footprint ~56x40 = 2240

// ---- CDNA5 async global->LDS (gfx1250, tracked by ASYNCcnt) --------------
__device__ __forceinline__ void async_ld_b128(unsigned lds_byte_off, const void* gaddr) {
    asm volatile("global_load_async_to_lds_b128 %0, %1, off"
                 :
                 : "v"(lds_byte_off), "v"(gaddr)
                 : "memory");
}

__device__ __forceinline__ void wait_async0() {
#if defined(__has_builtin) && __has_builtin(__builtin_amdgcn_s_wait_asynccnt)
    __builtin_amdgcn_s_wait_asynccnt(0);
#else
    asm volatile("s_wait_asynccnt 0" ::: "memory");
#endif
}

// Per-thread, per-angle projection geometry for one (x,y) column
struct Geo {
    float wu0, wu1, iv0, stepv;
    int u0, u1;
    int colvalid;
};

__device__ __forceinline__ Geo make_geo(float xc, float yc, float zc0, int a) {
    Geo g;
    const float theta = (float)a * (6.283185307179586f / 360.0f);
    float sn, cs;
    __sincosf(theta, &sn, &cs);
    const float xr  = xc * cs + yc * sn;
    const float yr  = -xc * sn + yc * cs;
    const float mag = 1536.0f / (1000.0f - xr);          // DSD/(DSO-xr), always > 0
    const float iu  = yr * mag * 0.5f + 127.5f;          // yr*mag/du + (U-1)/2
    g.colvalid = (iu >= 0.0f) && (iu <= 255.0f);
    const float iuc = fminf(fmaxf(iu, 0.0f), 255.0f);
    g.u0 = (int)floorf(iuc);
    g.u1 = min(g.u0 + 1, NU - 1);
    const float fu = iuc - (float)g.u0;
    g.wu0 = 1.0f - fu;
    g.wu1 = fu;
    g.stepv = mag;                                       // dz*mag/dv = mag (dz=dv=2)
    g.iv0   = zc0 * mag * 0.5f + 127.5f;
    return g;
}

// Block-reduce the detector bbox for this angle and asynchronously stage the
// rect into `buf`. mode: 0 = no valid column, 1 = staged in LDS, 2 = too big,
// use direct-global fallback in the compute phase.
__device__ __forceinline__ void bbox_and_stage(const Geo& g, const float* __restrict__ proj,
                                               size_t pbase, float* buf, int* s_box, int tid,
                                               int& mode, int& ulo4, int& vlo, int& w4) {
    mode = 0; ulo4 = 0; vlo = 0; w4 = 0;
    __syncthreads();                    // protect s_box / buf reuse
    if (tid == 0) { s_box[0] = INT_MAX; s_box[1] = -1; s_box[2] = INT_MAX; s_box[3] = -1; }
    __syncthreads();
    if (g.colvalid) {
        const float ivlo = fminf(fmaxf(g.iv0, 0.0f), 255.0f);
        const float ivhi = fminf(fmaxf(g.iv0 + (float)(ZCHUNK - 1) * g.stepv, 0.0f), 255.0f);
        atomicMin(&s_box[0], g.u0);
        atomicMax(&s_box[1], g.u1);
        atomicMin(&s_box[2], (int)floorf(ivlo));
        atomicMax(&s_box[3], min((int)floorf(ivhi) + 1, NV - 1));
    }
    __syncthreads();
    const int ulo = s_box[0], uhi = s_box[1], vhi = s_box[3];
    vlo = s_box[2];
    if (uhi < ulo) { mode = 0; return; }          // uniform

    // Align u-range to 4 floats so rows stage as 16B chunks (never crosses a
    // detector row: uhi <= 255 and 256 is a multiple of 4).
    ulo4 = ulo & ~3;
    w4   = (uhi - ulo4 + 4) >> 2;                 // row length in 16B chunks
    const int H = vhi - vlo + 1;
    if ((w4 << 2) * H > TILE_CAP) { mode = 2; return; }
    mode = 1;

    const unsigned lbase = (unsigned)(uintptr_t)(void*)buf;
    const float* rbase = proj + pbase + ulo4;
    const int nch = w4 * H;
    for (int k = tid; k < nch; k += 256) {
        const int r  = k / w4;
        const int cc = k - r * w4;
        async_ld_b128(lbase + 16u * (unsigned)k,
                      rbase + (size_t)(vlo + r) * NU + 4 * cc);
    }
}

// Bilinear accumulate for this angle into the per-thread z-column registers.
__device__ __forceinline__ void accumulate(const Geo& g, int mode, int ulo4, int vlo, int w4,
                                           const float* __restrict__ buf,
                                           const float* __restrict__ proj, size_t pbase,
                                           float acc[ZCHUNK]) {
    if (mode == 0 || !g.colvalid) return;
    float iv = g.iv0;
    if (mode == 1) {
        const int Wp  = w4 << 2;
        const int lu0 = g.u0 - ulo4;
        const int lu1 = g.u1 - ulo4;
#pragma unroll
        for (int j = 0; j < ZCHUNK; ++j) {
            const bool valid = (iv >= 0.0f) && (iv <= 255.0f);
            const float ivc = fminf(fmaxf(iv, 0.0f), 255.0f);
            const int   v0  = (int)floorf(ivc);
            const int   v1  = min(v0 + 1, NV - 1);
            const float fv  = ivc - (float)v0;
            const int r0 = (v0 - vlo) * Wp;
            const int r1 = (v1 - vlo) * Wp;
            const float q0 = buf[r0 + lu0] * g.wu0 + buf[r0 + lu1] * g.wu1;
            const float q1 = buf[r1 + lu0] * g.wu0 + buf[r1 + lu1] * g.wu1;
            acc[j] += valid ? (q0 + (q1 - q0) * fv) : 0.0f;
            iv += g.stepv;
        }
    } else {
#pragma unroll
        for (int j = 0; j < ZCHUNK; ++j) {
            const bool valid = (iv >= 0.0f) && (iv <= 255.0f);
            const float ivc = fminf(fmaxf(iv, 0.0f), 255.0f);
            const int   v0  = (int)floorf(ivc);
            const int   v1  = min(v0 + 1, NV - 1);
            const float fv  = ivc - (float)v0;
            const float p00 = proj[pbase + (size_t)v0 * NU + g.u0];
            const float p01 = proj[pbase + (size_t)v0 * NU + g.u1];
            const float p10 = proj[pbase + (size_t)v1 * NU + g.u0];
            const float p11 = proj[pbase + (size_t)v1 * NU + g.u1];
            const float q0 = p00 * g.wu0 + p01 * g.wu1;
            const float q1 = p10 * g.wu0 + p11 * g.wu1;
            acc[j] += valid ? (q0 + (q1 - q0) * fv) : 0.0f;
            iv += g.stepv;
        }
    }
}

// --------------------------------------------------------------------------
// Block = 16x16 (x,y) columns x ZCHUNK z-slices of one batch element.
// Software pipeline over angles with double-buffered async LDS staging:
//   iter a:  wait(ASYNCcnt==0) ; barrier          -> tile[a&1] complete
//            bbox+issue async loads for a+1       -> tile[(a+1)&1]
//            bilinear compute for a from tile[a&1]   (overlaps a+1 DMA)
// --------------------------------------------------------------------------
__global__ __launch_bounds__(256) void backproject_kernel(const float* __restrict__ proj,
                                                          float* __restrict__ out) {
    __shared__ __align__(16) float tile[2][TILE_CAP];
    __shared__ int s_box[4];

    const int tid = threadIdx.x;
    const int tx = tid & (TILE_X - 1);
    const int ty = tid >> 4;
    const int bx = blockIdx.x & 7;          // 128/16 = 8 tiles in x
    const int by = blockIdx.x >> 3;
    const int x  = bx * TILE_X + tx;
    const int y  = by * TILE_Y + ty;
    const int z0 = blockIdx.y * ZCHUNK;
    const int b  = blockIdx.z;

    const float xc  = ((float)x  - 63.5f) * 2.0f;
    const float yc  = ((float)y  - 63.5f) * 2.0f;
    const float zc0 = ((float)z0 - 63.5f) * 2.0f;

    float acc[ZCHUNK];
#pragma unroll
    for (int j = 0; j < ZCHUNK; ++j) acc[j] = 0.0f;

    // Prologue: stage angle 0 into tile[0]
    Geo    g_nxt  = make_geo(xc, yc, zc0, 0);
    size_t pb_nxt = (size_t)(b * NA) * (size_t)(NV * NU);
    int m_nxt, ulo_nxt, vlo_nxt, w4_nxt;
    bbox_and_stage(g_nxt, proj, pb_nxt, tile[0], s_box, tid, m_nxt, ulo_nxt, vlo_nxt, w4_nxt);

    for (int a = 0; a < NA; ++a) {
        wait_async0();          // this wave's staged loads for angle a are done
        __syncthreads();        // ... and every other wave's too

        const Geo    g    = g_nxt;
        const int    m    = m_nxt, ulo4 = ulo_nxt, vlo = vlo_nxt, w4 = w4_nxt;
        const size_t pb   = pb_nxt;

        if (a + 1 < NA) {       // issue next angle's DMA before computing this one
            g_nxt  = make_geo(xc, yc, zc0, a + 1);
            pb_nxt = (size_t)(b * NA + (a + 1)) * (size_t)(NV * NU);
            bbox_and_stage(g_nxt, proj, pb_nxt, tile[(a + 1) & 1], s_box, tid,
                           m_nxt, ulo_nxt, vlo_nxt, w4_nxt);
        }

        accumulate(g, m, ulo4, vlo, w4, tile[a & 1], proj, pb, acc);
    }

    // Write volume out[b, z, y, x] (coalesced in x), each voxel exactly once.
    float* outp = out + (((size_t)b * NZ + z0) * NY + y) * NX + x;
#pragma unroll
    for (int j = 0; j < ZCHUNK; ++j) {
        outp[(size_t)j * (NY * NX)] = acc[j];
    }
}

extern "C" void kernel_launch(void* const* d_in, const int* in_sizes, int n_in,
                              void* d_out, int out_size, void* d_ws, size_t ws_size,
                              hipStream_t stream) {
    (void)in_sizes; (void)n_in; (void)out_size; (void)d_ws; (void)ws_size;
    const float* proj = (const float*)d_in[0];   // [B,1,A,V,U] f32
    float* out = (float*)d_out;                  // [B,1,nz,ny,nx] f32

    dim3 grid((NX / TILE_X) * (NY / TILE_Y),     // 64 (x,y) tiles
              NZ / ZCHUNK,                       // 8 z-chunks
              NB);                               // 2 batches
    backproject_kernel<<<grid, 256, 0, stream>>>(proj, out);
}